// AdaptiveUnpooling_6828998000911
// MI455X (gfx1250) — compile-verified
//
#include <hip/hip_runtime.h>
#include <hip/hip_bf16.h>

// ---------------- problem constants (from reference) ----------------
#define NN        100000      // original_num_nodes
#define NP        50000       // pooled nodes
#define NC        64          // channels
#define NE        1250000     // edges
#define NDIR      (2 * NE)    // directed pairs

#define HASH_BITS 23
#define HASH_SIZE (1u << HASH_BITS)     // 8,388,608 entries (u64) = 64 MB, load <= ~0.3
#define HASH_MASK (HASH_SIZE - 1u)
#define HEMPTY    0xFFFFFFFFFFFFFFFFull

// workspace byte offsets
#define WS_FEAT   0                                  // NN*NC f32  = 25,600,000 B
#define WS_CNT    25600000                           // NN   u32   =    400,000 B
#define WS_TABLE  26000000                           // HASH u64   = 67,108,864 B (8-aligned)
#define WS_MISS   93108864                           // NN   u8    =    100,000 B

typedef __attribute__((ext_vector_type(2))) float v2f;
typedef __attribute__((ext_vector_type(8))) float v8f;

// ---------------------------------------------------------------
// Kernel 1: scatter pooled rows into x0 (= d_out) and clear "missing".
// One thread per 4 channels: NP*16 = 800,000 threads. 128-bit stores.
// ---------------------------------------------------------------
__global__ void scatter_kernel(const float* __restrict__ xa,
                               const int* __restrict__ perm,
                               float* __restrict__ out,
                               unsigned char* __restrict__ missing) {
    int tid = blockIdx.x * blockDim.x + threadIdx.x;
    if (tid >= NP * 16) return;
    int i = tid >> 4;            // pooled row
    int q = (tid & 15) * 4;      // channel base
    int p = perm[i];             // original slot
    const float4 v = *(const float4*)(xa + (size_t)i * NC + q);
    *(float4*)(out + (size_t)p * NC + q) = v;
    if (q == 0) missing[p] = 0;
}

// ---------------------------------------------------------------
// Kernel 2: dedup + scatter-add.
// Lane l of wave w owns directed pair id = w*32+l (NDIR % 32 == 0,
// NE % 32 == 16 -> guards are wave-uniform, the direction split is not).
// Phase 1: 32 parallel u64 hash-CAS inserts (L2-resident table).
// Phase 2: wave iterates ballot mask of fresh, non-self pairs and
// cooperatively gathers the 64-ch source row (2 ch/lane) and issues
// returnless global_atomic_add_f32 into feat_sum, + 1 count atomic.
// ---------------------------------------------------------------
__global__ void edge_kernel(const int* __restrict__ e0,
                            const int* __restrict__ e1,
                            const float* __restrict__ x0,     // = d_out after scatter
                            float* __restrict__ feat,
                            unsigned* __restrict__ cnt,
                            unsigned long long* __restrict__ table) {
    int id   = blockIdx.x * blockDim.x + threadIdx.x;
    int lane = threadIdx.x & 31;
    if (id >= NDIR) return;                       // whole-wave uniform

    int t, s;
    if (id < NE) { t = e0[id];      s = e1[id];      }
    else         { t = e1[id - NE]; s = e0[id - NE]; }

    // hide hash latency: start pulling the candidate source row toward L2/L0
    __builtin_prefetch(x0 + (size_t)s * NC, 0, 1);   // -> global_prefetch_b8

    bool fresh = false;
    if (t != s) {
        unsigned long long key =
            ((unsigned long long)(unsigned)t << 32) | (unsigned)s;
        unsigned long long h = key * 0x9E3779B97F4A7C15ull;
        unsigned slot = (unsigned)(h >> (64 - HASH_BITS));
        for (;;) {
            unsigned long long old = atomicCAS(&table[slot], HEMPTY, key);
            if (old == HEMPTY) { fresh = true;  break; }
            if (old == key)    { fresh = false; break; }
            slot = (slot + 1u) & HASH_MASK;
        }
    }

    unsigned long long mask = __ballot(fresh);     // wave32: low 32 bits
    while (mask) {
        int e = __ffsll((unsigned long long)mask) - 1;
        mask &= mask - 1ull;
        int tt = __shfl(t, e);
        int ss = __shfl(s, e);
        float v0 = x0[(size_t)ss * NC + lane];
        float v1 = x0[(size_t)ss * NC + lane + 32];
        atomicAdd(feat + (size_t)tt * NC + lane,      v0);
        atomicAdd(feat + (size_t)tt * NC + lane + 32, v1);
        if (lane == 0) atomicAdd(cnt + tt, 1u);
    }
}

// ---------------------------------------------------------------
// Kernel 3: WMMA epilogue.  out = fill ? feat*1/cnt : x0, done as
// D = S x Feat + C with S a masked diagonal, via 4 chained
// V_WMMA_F32_16X16X4_F32 per 16x16 tile (K split 16 = 4x4).
// One wave per tile; 6250x4 = 25000 tiles = 3125 full blocks,
// so EXEC is all-ones at every WMMA.
// ISA layouts (7.12.2):
//   A 16x4:  lane l -> M = l&15, VGPR v holds K = v + 2*(l>>4)
//   B 4x16:  lane l -> N = l&15, VGPR v holds K = v + 2*(l>>4)
//   C/D 16x16: lane l -> N = l&15, VGPR v holds M = v + 8*(l>>4)
// ---------------------------------------------------------------
__global__ void finalize_kernel(const float* __restrict__ feat,
                                const unsigned* __restrict__ cnt,
                                const unsigned char* __restrict__ missing,
                                float* __restrict__ out) {
    int wave = (blockIdx.x * blockDim.x + threadIdx.x) >> 5;
    int lane = threadIdx.x & 31;
    int rt = wave >> 2;                 // row-tile  [0, 6250)
    int ct = wave & 3;                  // col-tile  [0, 4)
    int rbase = rt * 16;
    int cbase = ct * 16;
    int lo = lane & 15;
    int hi = lane >> 4;                 // 0 or 1

    // diagonal scale for A's row m = lo
    int rowA = rbase + lo;
    unsigned cA = cnt[rowA];
    float scaleA = (missing[rowA] && cA > 0u) ? (1.0f / (float)cA) : 0.0f;

    // accumulator C = row-masked x0 tile (fill rows contribute 0)
    v8f c;
#pragma unroll
    for (int v = 0; v < 8; ++v) {
        int row = rbase + hi * 8 + v;
        float x0v = out[(size_t)row * NC + cbase + lo];
        bool fill = missing[row] && (cnt[row] > 0u);
        c[v] = fill ? 0.0f : x0v;
    }

    // D = sum_j A_j(16x4 diag slice) x B_j(4x16 feat rows) + C
#pragma unroll
    for (int j = 0; j < 4; ++j) {
        v2f a, b;
        int k0 = 2 * hi;                               // k of VGPR 0 in this lane-half
        a.x = (lo == j * 4 + k0)     ? scaleA : 0.0f;  // A[m][k0]
        a.y = (lo == j * 4 + k0 + 1) ? scaleA : 0.0f;  // A[m][k0+1]
        b.x = feat[(size_t)(rbase + j * 4 + k0)     * NC + cbase + lo];  // B[k0][n]
        b.y = feat[(size_t)(rbase + j * 4 + k0 + 1) * NC + cbase + lo];  // B[k0+1][n]
        c = __builtin_amdgcn_wmma_f32_16x16x4_f32(
                /*neg_a=*/false, a, /*neg_b=*/false, b,
                /*c_mod=*/(short)0, c, /*reuse_a=*/false, /*reuse_b=*/false);
    }

    // store D (same addressing as C tile; tile owned exclusively by this wave)
#pragma unroll
    for (int v = 0; v < 8; ++v) {
        int row = rbase + hi * 8 + v;
        out[(size_t)row * NC + cbase + lo] = c[v];
    }
}

// ---------------------------------------------------------------
extern "C" void kernel_launch(void* const* d_in, const int* in_sizes, int n_in,
                              void* d_out, int out_size, void* d_ws, size_t ws_size,
                              hipStream_t stream) {
    const float* xa   = (const float*)d_in[0];      // [NP, NC]
    const int*   perm = (const int*)d_in[1];        // [NP]
    const int*   e0   = (const int*)d_in[2];        // edge_index[0], [NE]
    const int*   e1   = e0 + NE;                    // edge_index[1], [NE]
    float*       out  = (float*)d_out;              // [NN, NC]

    char* ws = (char*)d_ws;
    float*              featsum = (float*)(ws + WS_FEAT);
    unsigned*           cnt     = (unsigned*)(ws + WS_CNT);
    unsigned long long* table   = (unsigned long long*)(ws + WS_TABLE);
    unsigned char*      missing = (unsigned char*)(ws + WS_MISS);

    // re-initialize all state every call (graph-capture-safe memset nodes)
    hipMemsetAsync(featsum, 0,    (size_t)NN * NC * sizeof(float), stream);
    hipMemsetAsync(cnt,     0,    (size_t)NN * sizeof(unsigned), stream);
    hipMemsetAsync(table,   0xFF, (size_t)HASH_SIZE * sizeof(unsigned long long), stream);
    hipMemsetAsync(missing, 1,    (size_t)NN, stream);
    hipMemsetAsync(out,     0,    (size_t)NN * NC * sizeof(float), stream);

    // 1) scatter pooled rows into x0 (= out)
    {
        int threads = NP * 16;                       // 800,000
        scatter_kernel<<<(threads + 255) / 256, 256, 0, stream>>>(xa, perm, out, missing);
    }
    // 2) dedup + scatter-add (2.5M directed pairs, 1 lane each)
    {
        int threads = NDIR;                          // 2,500,000 (multiple of 32)
        edge_kernel<<<(threads + 255) / 256, 256, 0, stream>>>(e0, e1, out, featsum, cnt, table);
    }
    // 3) WMMA masked-diagonal epilogue: 25,000 tiles, 8 waves/block -> 3125 blocks
    {
        finalize_kernel<<<3125, 256, 0, stream>>>(featsum, cnt, missing, out);
    }
}